// YOLOv2Loss_37778532336464
// MI455X (gfx1250) — compile-verified
//
#include <hip/hip_runtime.h>
#include <hip/hip_bf16.h>

// YOLOv2 loss for gfx1250 (MI455X).
// Structure:
//   k_zero    : out[0] = 0
//   k_targets : 512 threads, one per (b,g): decode assigned pred box, IoU vs its GT
//               (tobj), target record {tx,ty,gw,gh,tobj,cls} into d_ws (16 KB).
//   k_loss    : one thread per (b,a,y,x) cell (216,320 cells). Reads channels 0..4
//               only (4.3 MB dense); class channels gathered only at <=512 positive
//               cells. noobj test is division-free. Per-wave reduction via
//               v_wmma_f32_16x16x4_f32 (ones-vector row-sum trick) + shfl_xor +
//               atomicAdd.

#define W52   52
#define HW    2704          // 52*52
#define NB    16
#define NANCH 5
#define CPA   85
#define NG    32
#define NCLS  80

typedef __attribute__((ext_vector_type(2))) float v2f;
typedef __attribute__((ext_vector_type(8))) float v8f;

__device__ __forceinline__ float sigmoidf_(float x) {
  return 1.f / (1.f + __expf(-x));
}

// Wave32 sum reduction using V_WMMA_F32_16X16X4_F32.
// A (16x4 f32): lane L<16 supplies {A[L][0], A[L][1]}, lane L>=16 supplies
// {A[L-16][2], A[L-16][3]}  (ISA 7.12.2). Set A[m][0]=acc_m, A[m][2]=acc_{m+16},
// other K zero; B = ones(4x16)  =>  D[m][n] = acc_m + acc_{m+16} for every n.
// D layout: lane n (0..15) VGPR r = D[r][n]; lane n+16 VGPR r = D[8+r][n].
// Per-lane sum of the 8 D regs gives half the total; shfl_xor(16) completes it.
__device__ __forceinline__ float wave_sum32(float v) {
#if __has_builtin(__builtin_amdgcn_wmma_f32_16x16x4_f32)
  v2f a;  a[0] = v;   a[1] = 0.f;
  v2f b1; b1[0] = 1.f; b1[1] = 1.f;
  v8f c = {0.f, 0.f, 0.f, 0.f, 0.f, 0.f, 0.f, 0.f};
  c = __builtin_amdgcn_wmma_f32_16x16x4_f32(
      /*neg_a=*/false, a, /*neg_b=*/false, b1,
      /*c_mod=*/(short)0, c, /*reuse_a=*/false, /*reuse_b=*/false);
  float s = ((c[0] + c[1]) + (c[2] + c[3])) + ((c[4] + c[5]) + (c[6] + c[7]));
  s += __shfl_xor(s, 16, 32);
  return s;
#else
  #pragma unroll
  for (int off = 16; off; off >>= 1) v += __shfl_xor(v, off, 32);
  return v;
#endif
}

__global__ void k_zero(float* out) { out[0] = 0.f; }

__global__ __launch_bounds__(256) void k_targets(
    const float* __restrict__ pred, const float* __restrict__ anchors,
    const float* __restrict__ gt_boxes, const int* __restrict__ gt_cls,
    const int* __restrict__ gt_anchor, float* __restrict__ rec) {
  int i = blockIdx.x * blockDim.x + threadIdx.x;   // (b,g) flat, 0..511
  if (i >= NB * NG) return;
  int b = i >> 5;
  const float* gb = gt_boxes + i * 4;
  float cx = gb[0], cy = gb[1], gw = gb[2], gh = gb[3];
  int xi = (int)(cx * 52.f);
  int yi = (int)(cy * 52.f);
  int ai = gt_anchor[i];
  float aw = anchors[2 * ai], ah = anchors[2 * ai + 1];
  int base = ((b * 425 + ai * CPA) * W52 + yi) * W52 + xi;
  float bx = sigmoidf_(pred[base]);
  float by = sigmoidf_(pred[base + HW]);
  float bw = __expf(pred[base + 2 * HW]) * aw;
  float bh = __expf(pred[base + 3 * HW]) * ah;
  // IoU(pred box, its GT) -> tobj
  float ax0 = bx - 0.5f * bw, ay0 = by - 0.5f * bh;
  float ax1 = bx + 0.5f * bw, ay1 = by + 0.5f * bh;
  float gx0 = cx - 0.5f * gw, gy0 = cy - 0.5f * gh;
  float gx1 = cx + 0.5f * gw, gy1 = cy + 0.5f * gh;
  float iw = fmaxf(fminf(ax1, gx1) - fmaxf(ax0, gx0), 0.f);
  float ih = fmaxf(fminf(ay1, gy1) - fmaxf(ay0, gy0), 0.f);
  float inter = iw * ih;
  float iou = inter / (bw * bh + gw * gh - inter);
  float* r = rec + i * 8;
  r[0] = fmodf(cx, 1.f / 52.f);
  r[1] = fmodf(cy, 1.f / 52.f);
  r[2] = gw;
  r[3] = gh;
  r[4] = iou;
  r[5] = (float)gt_cls[i];
  r[6] = 0.f;
  r[7] = 0.f;
}

__global__ __launch_bounds__(256) void k_loss(
    const float* __restrict__ pred, const float* __restrict__ anchors,
    const float* __restrict__ gt_boxes, const int* __restrict__ gt_anchor,
    const int* __restrict__ seen_p, const float* __restrict__ rec,
    float* __restrict__ out) {
  __shared__ float sx0[NG], sy0[NG], sx1[NG], sy1[NG], sB[NG];  // sB = 0.6*areaB
  __shared__ int sxi[NG], syi[NG], sai[NG];

  int ba = blockIdx.y;            // b*NA + a
  int b = ba / NANCH;
  int a = ba - b * NANCH;
  int tid = threadIdx.x;

  if (tid < NG) {
    const float* gb = gt_boxes + (b * NG + tid) * 4;
    float cx = gb[0], cy = gb[1], w = gb[2], h = gb[3];
    sx0[tid] = cx - 0.5f * w;  sy0[tid] = cy - 0.5f * h;
    sx1[tid] = cx + 0.5f * w;  sy1[tid] = cy + 0.5f * h;
    sB[tid]  = 0.6f * w * h;
    sxi[tid] = (int)(cx * 52.f);
    syi[tid] = (int)(cy * 52.f);
    sai[tid] = gt_anchor[b * NG + tid];
  }
  __syncthreads();

  float aw = anchors[2 * a], ah = anchors[2 * a + 1];
  int seen = seen_p[0];
  int cell = blockIdx.x * blockDim.x + tid;   // within (b,a) slice, 0..2703
  float acc = 0.f;

  if (cell < HW) {
    int y = cell / W52, x = cell - y * W52;
    int base = ((b * 425 + a * CPA) * W52 + y) * W52 + x;
    float bx   = sigmoidf_(pred[base]);
    float by   = sigmoidf_(pred[base + HW]);
    float bw   = __expf(pred[base + 2 * HW]) * aw;
    float bh   = __expf(pred[base + 3 * HW]) * ah;
    float conf = sigmoidf_(pred[base + 4 * HW]);

    float ax0 = bx - 0.5f * bw, ay0 = by - 0.5f * bh;
    float ax1 = bx + 0.5f * bw, ay1 = by + 0.5f * bh;
    float sA = 0.6f * bw * bh;

    bool high = false;   // any IoU > 0.6 (division-free: 1.6*inter > 0.6*(A+B))
    int posg = -1;       // winning GT assigned to this (a,y,x), if any
    #pragma unroll 8
    for (int g = 0; g < NG; ++g) {
      float iw = fmaxf(fminf(ax1, sx1[g]) - fmaxf(ax0, sx0[g]), 0.f);
      float ih = fmaxf(fminf(ay1, sy1[g]) - fmaxf(ay0, sy0[g]), 0.f);
      float inter = iw * ih;
      high = high || (1.6f * inter > sA + sB[g]);
      if (sai[g] == a && syi[g] == y && sxi[g] == x) posg = g;
    }
    bool pos = posg >= 0;

    if (!high && !pos) acc += conf * conf;                // noobj loss
    if (!pos && seen < 12800) {                           // prior loss (x0.01)
      float e0 = bx - (0.5f / 52.f), e1 = by - (0.5f / 52.f);
      float e2 = bw - aw, e3 = bh - ah;
      acc += 0.01f * (e0 * e0 + e1 * e1 + e2 * e2 + e3 * e3);
    }
    if (pos) {
      const float* r = rec + (b * NG + posg) * 8;
      float d0 = bx - r[0], d1 = by - r[1], d2 = bw - r[2], d3 = bh - r[3];
      acc += d0 * d0 + d1 * d1 + d2 * d2 + d3 * d3;       // box loss
      float dob = conf - r[4];
      acc += 5.f * dob * dob;                             // obj loss (x5)
      int cls = (int)r[5];
      int cb = base + 5 * HW;
      __builtin_prefetch(pred + cb, 0, 1);                // global_prefetch_b8
      // softmax loss = sum_c s_c^2 - 2*s_k + 1  (t one-hot at cls)
      float m = -3.402823e38f;
      for (int c = 0; c < NCLS; ++c) m = fmaxf(m, pred[cb + c * HW]);
      float S = 0.f, Q = 0.f, ek = 0.f;
      for (int c = 0; c < NCLS; ++c) {
        float e = __expf(pred[cb + c * HW] - m);
        S += e;
        Q += e * e;
        if (c == cls) ek = e;
      }
      float inv = 1.f / S;
      acc += (Q * inv * inv - 2.f * ek * inv) + 1.f;      // cls loss (positive)
    } else {
      acc += 1.f / 80.f;                                  // cls loss (softmax of 0s)
    }
  }

#if __has_builtin(__builtin_amdgcn_s_wait_tensorcnt)
  __builtin_amdgcn_s_wait_tensorcnt(0);                   // CDNA5 split-counter path
#endif

  // Full waves everywhere (blockDim=256): EXEC all-ones at the WMMA.
  float wsum = wave_sum32(acc);
  if ((tid & 31) == 0) atomicAdd(out, wsum);
}

extern "C" void kernel_launch(void* const* d_in, const int* in_sizes, int n_in,
                              void* d_out, int out_size, void* d_ws, size_t ws_size,
                              hipStream_t stream) {
  const float* pred      = (const float*)d_in[0];
  const float* anchors   = (const float*)d_in[1];
  const float* gt_boxes  = (const float*)d_in[2];
  const int*   gt_cls    = (const int*)d_in[3];
  const int*   gt_anchor = (const int*)d_in[4];
  const int*   seen      = (const int*)d_in[5];
  float* out = (float*)d_out;
  float* rec = (float*)d_ws;                     // 512 * 8 floats = 16 KB

  hipLaunchKernelGGL(k_zero, dim3(1), dim3(1), 0, stream, out);
  hipLaunchKernelGGL(k_targets, dim3(2), dim3(256), 0, stream,
                     pred, anchors, gt_boxes, gt_cls, gt_anchor, rec);
  dim3 grid(11, NB * NANCH);                     // 11*256 >= 2704 cells per (b,a)
  hipLaunchKernelGGL(k_loss, grid, dim3(256), 0, stream,
                     pred, anchors, gt_boxes, gt_anchor, seen, rec, out);
}